// StreamingPCENTransform_68667937128904
// MI455X (gfx1250) — compile-verified
//
#include <hip/hip_runtime.h>

// PCEN: x[64,8192,80] f32 -> pcen, EMA scan over T then elementwise power law.
// Chunked scan (16 chunks of 512) + intra-chunk 16x16 scan-as-matmul via
// V_WMMA_F32_16X16X4_F32 (4 accumulating K=4 WMMAs = K=16 contraction).

typedef __attribute__((ext_vector_type(2))) float v2f;
typedef __attribute__((ext_vector_type(8))) float v8f;

#define B_     64
#define T_     8192
#define F_     80
#define FT_    (F_ / 16)      // 5 feature tiles
#define NC_    16             // time chunks
#define L_     (T_ / NC_)     // 512 steps per chunk
#define NBLK_  (L_ / 16)      // 32 wmma blocks per chunk

#define S_     0.025f
#define AA_    0.975f         // 1 - S
#define ALPHA_ 0.98f
#define EPS_   1e-6f
#define SQRT2_ 1.41421356237309515f

// ---------------------------------------------------------------------------
// Pass 1: per-(b, chunk, f) zero-state chunk contribution
//   c_k = recurrence over the chunk with zero initial state.
// Streaming, coalesced over f; prefetch ahead to keep HBM busy.
// ---------------------------------------------------------------------------
__global__ void pcen_chunk_carry(const float* __restrict__ x,
                                 float* __restrict__ c) {
    int idx = blockIdx.x * blockDim.x + threadIdx.x;   // B*NC*F
    if (idx >= B_ * NC_ * F_) return;
    int f = idx % F_;
    int k = (idx / F_) % NC_;
    int b = idx / (F_ * NC_);
    const float* p = x + ((size_t)b * T_ + (size_t)k * L_) * F_ + f;
    float m = 0.f;
    #pragma unroll 4
    for (int j = 0; j < L_; ++j) {
        if ((j & 15) == 0)
            __builtin_prefetch(p + (size_t)(j + 64) * F_, 0, 1); // global_prefetch_b8
        m = fmaf(AA_, m, S_ * p[(size_t)j * F_]);
    }
    c[idx] = m;
}

// ---------------------------------------------------------------------------
// Pass 2: tiny per-series scan across the 16 chunk carries.
// carry_in[b,0,f] = x[b,0,f]  (exact: since a=1-s, a*x0 + s*x0 = x0 = M[0])
// carry_in[b,k,f] = a^L * carry_in[b,k-1,f] + c[b,k-1,f]
// ---------------------------------------------------------------------------
__global__ void pcen_chunk_scan(const float* __restrict__ x,
                                const float* __restrict__ c,
                                float* __restrict__ cin) {
    int idx = blockIdx.x * blockDim.x + threadIdx.x;   // B*F
    if (idx >= B_ * F_) return;
    int f = idx % F_;
    int b = idx / F_;
    const float aL = __powf(AA_, (float)L_);           // a^512
    float m = x[(size_t)b * T_ * F_ + f];
    cin[((size_t)b * NC_ + 0) * F_ + f] = m;
    for (int k = 1; k < NC_; ++k) {
        m = fmaf(aL, m, c[((size_t)b * NC_ + (k - 1)) * F_ + f]);
        cin[((size_t)b * NC_ + k) * F_ + f] = m;
    }
}

// ---------------------------------------------------------------------------
// Pass 3: one wave32 per (b, chunk, 16-feature tile).
// Per 16-step block:  M[t,n] = a^{t+1} * m_prev[n] + sum_k W[t,k] x[k,n]
// where W[t,k] = S * a^{t-k} (lower triangular, constant) -> D = A*B + C with
// V_WMMA_F32_16X16X4_F32, 4 accumulating K=4 slices. Carry folded into C.
// Row t=15 of D (VGPR7, lanes 16..31) broadcast via ds_bpermute for the next
// block's carry. PCEN tail uses raw v_log/v_exp/v_sqrt (inputs are provably
// in the fast range: M+eps in [1e-6,~1] => |log2| <= 20; sqrt arg >= 2).
// ---------------------------------------------------------------------------
__global__ void __launch_bounds__(128)
pcen_wmma_main(const float* __restrict__ x,
               const float* __restrict__ cin,
               float* __restrict__ out) {
    const int lane = threadIdx.x & 31;
    const int wid  = blockIdx.x * (blockDim.x >> 5) + (threadIdx.x >> 5);
    const int ft = wid % FT_;
    const int k  = (wid / FT_) % NC_;
    const int b  = wid / (FT_ * NC_);

    const int  n     = lane & 15;        // N column (feature within tile)
    const bool hi    = lane >= 16;
    const int  fbase = ft * 16;
    const int  t_m   = lane & 15;        // A-matrix M row held by this lane

    // A-matrix (16x4 f32, 2 VGPRs): lane holds M=lane&15; vgpr v holds
    // K = 4j + (hi?2:0) + v.  W[t][kk] = S * a^(t-kk) for kk<=t else 0.
    v2f a_sl[4];
    #pragma unroll
    for (int j = 0; j < 4; ++j)
        #pragma unroll
        for (int v = 0; v < 2; ++v) {
            int kk = 4 * j + (hi ? 2 : 0) + v;
            a_sl[j][v] = (kk <= t_m) ? S_ * __powf(AA_, (float)(t_m - kk)) : 0.f;
        }

    const float hfac = hi ? __powf(AA_, 8.f) : 1.f;    // extra a^8 for M rows 8..15
    float arow[8];                                     // a^{r+1}
    {
        float av = AA_;
        #pragma unroll
        for (int r = 0; r < 8; ++r) { arow[r] = av; av *= AA_; }
    }

    // carry-in for this chunk, broadcast per feature column to both lane halves
    float m_prev = cin[((size_t)b * NC_ + k) * F_ + fbase + n];

    const size_t base = ((size_t)b * T_ + (size_t)k * L_) * F_;

    for (int blk = 0; blk < NBLK_; ++blk) {
        const int t0 = blk * 16;

        // C init = carry term a^{t+1} * m_prev  (t = r + 8*hi)
        v8f cacc;
        #pragma unroll
        for (int r = 0; r < 8; ++r) cacc[r] = arow[r] * hfac * m_prev;

        // K=16 contraction as 4 accumulating K=4 WMMAs
        #pragma unroll
        for (int j = 0; j < 4; ++j) {
            const int kk0 = 4 * j + (hi ? 2 : 0);
            v2f bsl;
            bsl[0] = x[base + (size_t)(t0 + kk0 + 0) * F_ + fbase + n];
            bsl[1] = x[base + (size_t)(t0 + kk0 + 1) * F_ + fbase + n];
            cacc = __builtin_amdgcn_wmma_f32_16x16x4_f32(
                false, a_sl[j], false, bsl, (short)0, cacc, false, false);
        }

        // next-block carry: row M=15 lives in cacc[7] on lanes 16..31
        {
            int src = __float_as_int(cacc[7]);
            int idx = (16 + n) << 2;                   // byte index for bpermute
            m_prev = __int_as_float(__builtin_amdgcn_ds_bpermute(idx, src));
        }

        // PCEN elementwise + store: element (t = t0 + r + 8*hi, n)
        // (M+eps)^-alpha = exp2(-alpha * log2(M+eps)); raw trans ops, no fixups.
        #pragma unroll
        for (int r = 0; r < 8; ++r) {
            const int t = t0 + r + (hi ? 8 : 0);
            const size_t off = base + (size_t)t * F_ + fbase + n;
            const float xv   = x[off];                 // L0/L2-hot reload
            const float mv   = cacc[r];
            const float pinv = __builtin_amdgcn_exp2f(
                                   -ALPHA_ * __builtin_amdgcn_logf(mv + EPS_));
            out[off] = __builtin_amdgcn_sqrtf(fmaf(xv, pinv, 2.0f)) - SQRT2_;
        }
    }
}

// ---------------------------------------------------------------------------
extern "C" void kernel_launch(void* const* d_in, const int* in_sizes, int n_in,
                              void* d_out, int out_size, void* d_ws, size_t ws_size,
                              hipStream_t stream) {
    const float* x   = (const float*)d_in[0];
    float*       out = (float*)d_out;
    float*       c   = (float*)d_ws;                        // [B][NC][F]
    float*       cin = c + (size_t)B_ * NC_ * F_;           // [B][NC][F]

    {
        int nthr = B_ * NC_ * F_;
        pcen_chunk_carry<<<(nthr + 255) / 256, 256, 0, stream>>>(x, c);
    }
    {
        int nthr = B_ * F_;
        pcen_chunk_scan<<<(nthr + 255) / 256, 256, 0, stream>>>(x, c, cin);
    }
    {
        // one wave per (b, chunk, ftile): 64*16*5 = 5120 waves, 4 waves/block
        int nwaves = B_ * NC_ * FT_;
        pcen_wmma_main<<<nwaves / 4, 128, 0, stream>>>(x, cin, out);
    }
}